// ProposalLayer_6493990552184
// MI455X (gfx1250) — compile-verified
//
#include <hip/hip_runtime.h>
#include <stdint.h>

// ---------------- Problem constants (match reference) ----------------
#define BATCH      8
#define NANCH      262144
#define PRE        6000        // PRE_NMS_LIMIT
#define PROP       1000        // PROPOSAL_COUNT
#define BINS       4096
#define NEG_INF    (-1e30f)
#define NMS_T      0.7f

// ---------------- Workspace layout (32-bit word units) ----------------
// [0, BATCH*BINS)            : int   histogram
// [BATCH*BINS, +64)          : int   meta  (kb, A per batch)
// [CAND_OFF ...)             : float candidate SoA: y1,x1,y2,x2,score
#define HIST_INTS   (BATCH * BINS)          // 32768
#define META_INTS   64
#define CAND_OFF    (HIST_INTS + META_INTS) // 32832 words -> byte offset 131328 (16B aligned)
#define CAND_STRIDE (BATCH * PRE)           // 48000 floats per array

typedef unsigned int v4u __attribute__((ext_vector_type(4)));
typedef int          v8i __attribute__((ext_vector_type(8)));
typedef int          v4i __attribute__((ext_vector_type(4)));

// ---------------- CDNA5-specific helpers ----------------
__device__ __forceinline__ void async_copy_b128(const float* gsrc, float* ldst) {
  // CDNA5 async global->LDS (tracked by ASYNCcnt).
  uint64_t ga = (uint64_t)(uintptr_t)gsrc;
  uint32_t la = (uint32_t)(uintptr_t)ldst;
  asm volatile("global_load_async_to_lds_b128 %0, %1, off"
               :: "v"(la), "v"(ga) : "memory");
}

__device__ __forceinline__ void wait_async0() {
#if __has_builtin(__builtin_amdgcn_s_wait_asynccnt)
  __builtin_amdgcn_s_wait_asynccnt(0);
#else
  asm volatile("s_wait_asynccnt 0" ::: "memory");
#endif
}

__device__ __forceinline__ void wait_tensor0() {
#if __has_builtin(__builtin_amdgcn_s_wait_tensorcnt)
  __builtin_amdgcn_s_wait_tensorcnt(0);
#else
  asm volatile("s_wait_tensorcnt 0" ::: "memory");
#endif
}

// =====================================================================
// Kernel 1: per-batch 4096-bin histogram of objectness scores.
// grid = BATCH*NANCH/4096 blocks of 256; each block covers 4096 items of one batch.
// =====================================================================
__global__ __launch_bounds__(256) void hist_kernel(const float* __restrict__ probs,
                                                   int* __restrict__ hist) {
  __shared__ int lh[BINS];
  const int tid = threadIdx.x;
  for (int i = tid; i < BINS; i += 256) lh[i] = 0;
  __syncthreads();

  const long long base = (long long)blockIdx.x * 4096;
  const int b = (int)(base / NANCH);
  #pragma unroll 4
  for (int j = 0; j < 16; ++j) {
    const long long idx = base + tid + (long long)j * 256;
    // speculative streaming prefetch of a future chunk (gfx1250 global_prefetch_b8)
    if (idx + 8192 < (long long)BATCH * NANCH)
      __builtin_prefetch(&probs[(idx + 8192) * 2 + 1], 0, 0);
    const float s = probs[idx * 2 + 1];
    int bin = (int)(s * (float)BINS);
    bin = bin < 0 ? 0 : (bin > BINS - 1 ? BINS - 1 : bin);
    atomicAdd(&lh[bin], 1);
  }
  __syncthreads();
  for (int i = tid; i < BINS; i += 256) {
    const int c = lh[i];
    if (c) atomicAdd(&hist[b * BINS + i], c);
  }
}

// =====================================================================
// Kernel 2: suffix scan of histogram -> threshold bin kb and strictly-above count A.
// =====================================================================
__global__ __launch_bounds__(1024) void thresh_kernel(const int* __restrict__ hist,
                                                      int* __restrict__ meta) {
  __shared__ int cum[BINS];
  const int b = blockIdx.x, tid = threadIdx.x;
  for (int i = tid; i < BINS; i += 1024) cum[i] = hist[b * BINS + i];
  __syncthreads();
  for (int d = 1; d < BINS; d <<= 1) {
    int v[4];
    #pragma unroll
    for (int k = 0; k < 4; ++k) {
      const int i = tid + k * 1024;
      v[k] = cum[i] + ((i + d < BINS) ? cum[i + d] : 0);
    }
    __syncthreads();
    #pragma unroll
    for (int k = 0; k < 4; ++k) cum[tid + k * 1024] = v[k];
    __syncthreads();
  }
  #pragma unroll
  for (int k = 0; k < 4; ++k) {
    const int i = tid + k * 1024;
    const int c  = cum[i];
    const int cn = (i + 1 < BINS) ? cum[i + 1] : 0;
    if (c >= PRE && cn < PRE) { meta[b * 2 + 0] = i; meta[b * 2 + 1] = cn; }
  }
  if (tid == 0 && cum[0] < PRE) {
    meta[b * 2 + 0] = 0;
    meta[b * 2 + 1] = cum[1];
  }
}

// =====================================================================
// Kernel 3: deterministic stream compaction + box decode.
// =====================================================================
__global__ __launch_bounds__(1024) void compact_kernel(const float* __restrict__ probs,
                                                       const float* __restrict__ bbox,
                                                       const float* __restrict__ anchors,
                                                       const int* __restrict__ meta,
                                                       float* __restrict__ cand) {
  const int b = blockIdx.x;
  const int tid = threadIdx.x;
  const int lane = tid & 31, wav = tid >> 5;
  __shared__ int wA[32], wT[32], sA[33], sT[33];
  __shared__ int baseA, baseT;
  if (tid == 0) { baseA = 0; baseT = 0; }

  const int kb   = meta[b * 2 + 0];
  const int A    = meta[b * 2 + 1];
  const int need = PRE - A;

  float* cy1 = cand + 0 * CAND_STRIDE + b * PRE;
  float* cx1 = cand + 1 * CAND_STRIDE + b * PRE;
  float* cy2 = cand + 2 * CAND_STRIDE + b * PRE;
  float* cx2 = cand + 3 * CAND_STRIDE + b * PRE;
  float* cs  = cand + 4 * CAND_STRIDE + b * PRE;
  __syncthreads();

  for (int base0 = 0; base0 < NANCH; base0 += 1024) {
    const long long n = (long long)b * NANCH + base0 + tid;
    const float s = probs[n * 2 + 1];
    int bin = (int)(s * (float)BINS);
    bin = bin < 0 ? 0 : (bin > BINS - 1 ? BINS - 1 : bin);
    const bool fa = (bin > kb);
    const bool ft = (bin == kb);

    const unsigned long long mA = __ballot(fa);
    const unsigned long long mT = __ballot(ft);
    const unsigned long long lt = (1ull << lane) - 1ull;
    const int pA = __popcll(mA & lt);
    const int pT = __popcll(mT & lt);
    if (lane == 0) { wA[wav] = __popcll(mA); wT[wav] = __popcll(mT); }
    __syncthreads();
    if (tid == 0) {
      int accA = baseA, accT = baseT;
      for (int w = 0; w < 32; ++w) { sA[w] = accA; accA += wA[w]; sT[w] = accT; accT += wT[w]; }
      sA[32] = accA; sT[32] = accT;
    }
    __syncthreads();

    int slot = -1;
    if (fa) slot = sA[wav] + pA;
    else if (ft) { const int t = sT[wav] + pT; if (t < need) slot = A + t; }

    if (slot >= 0 && slot < PRE) {
      const float d0 = bbox[n * 4 + 0] * 0.1f, d1 = bbox[n * 4 + 1] * 0.1f;
      const float d2 = bbox[n * 4 + 2] * 0.2f, d3 = bbox[n * 4 + 3] * 0.2f;
      const float a0 = anchors[n * 4 + 0], a1 = anchors[n * 4 + 1];
      const float a2 = anchors[n * 4 + 2], a3 = anchors[n * 4 + 3];
      float h = a2 - a0, w = a3 - a1;
      const float cy = a0 + 0.5f * h + d0 * h;
      const float cx = a1 + 0.5f * w + d1 * w;
      h *= expf(d2);  w *= expf(d3);
      float y1 = cy - 0.5f * h, x1 = cx - 0.5f * w;
      float y2 = y1 + h,        x2 = x1 + w;
      y1 = fminf(fmaxf(y1, 0.f), 1.f);  x1 = fminf(fmaxf(x1, 0.f), 1.f);
      y2 = fminf(fmaxf(y2, 0.f), 1.f);  x2 = fminf(fmaxf(x2, 0.f), 1.f);
      cy1[slot] = y1; cx1[slot] = x1; cy2[slot] = y2; cx2[slot] = x2; cs[slot] = s;
    }
    __syncthreads();
    if (tid == 0) { baseA = sA[32]; baseT = sT[32]; }
    __syncthreads();
  }

  const int tieTaken = (baseT < need) ? baseT : need;
  const int placed = A + tieTaken;
  for (int i = placed + tid; i < PRE; i += 1024) {
    cy1[i] = 0.f; cx1[i] = 0.f; cy2[i] = 0.f; cx2[i] = 0.f; cs[i] = NEG_INF;
  }
}

// =====================================================================
// Kernel 4: greedy NMS, one 1024-thread block per batch image.
// Box rows staged via Tensor Data Mover (2D tile: 4 rows x 6000 f32, row pitch
// 48000 f32 -> contiguous LDS); score row staged via async-to-LDS b128.
// Then registers (6 candidates/thread), wave32 shuffle argmax, 1000 rounds.
// =====================================================================
__global__ __launch_bounds__(1024) void nms_kernel(const float* __restrict__ cand,
                                                   float* __restrict__ out) {
  const int b = blockIdx.x, tid = threadIdx.x;
  const int lane = tid & 31, wav = tid >> 5;

  __shared__ __align__(16) float stage[4 * PRE];  // rows: y1,x1,y2,x2 (TDM dest)
  __shared__ __align__(16) float sS[PRE];
  __shared__ float rS[32];
  __shared__ int   rI[32];
  __shared__ float bestBox[4];
  __shared__ float bestS_sh;
  __shared__ int   bestI_sh;
  if (tid == 0) { bestBox[0] = bestBox[1] = bestBox[2] = bestBox[3] = 0.f; }

  const float* gbase = cand + (size_t)b * PRE;     // y1 row; rows CAND_STRIDE apart

#if __has_builtin(__builtin_amdgcn_tensor_load_to_lds)
  // ---- TDM: one DMA for the 4 box rows (TENSORcnt path) ----
  // This toolchain exposes the 6-arg form:
  //   (uint32x4 g0, int32x8 g1, int32x4 g2, int32x4 g3, int32x8 extra, i32 cpol)
  if (tid == 0) {
    const uint64_t ga = (uint64_t)(uintptr_t)gbase;
    const uint32_t la = (uint32_t)(uintptr_t)&stage[0];
    const uint32_t TD0    = CAND_STRIDE;   // tensor_dim0 (elements)
    const uint32_t TD1    = 5u;            // tensor_dim1 (rows in tensor)
    const uint32_t TILE0  = PRE;           // tile_dim0 = 6000
    const uint32_t TILE1  = 4u;            // tile_dim1 = 4 rows
    const uint64_t STRIDE0 = CAND_STRIDE;  // tensor_dim0_stride (elements)
    v4u g0 = { 1u,                                             // count=1 (valid)
               la,                                             // lds_addr [63:32]
               (uint32_t)(ga & 0xffffffffull),                 // global_addr lo
               (uint32_t)((ga >> 32) & 0x01ffffffull) | (2u << 30) }; // addr hi + type=2
    v8i g1 = { (int)(2u << 16),                                // data_size=4B
               (int)((TD0 & 0xffffu) << 16),                   // tensor_dim0 lo16
               (int)((TD0 >> 16) | ((TD1 & 0xffffu) << 16)),   // dim0 hi16 | dim1 lo16
               (int)((TD1 >> 16) | (TILE0 << 16)),             // dim1 hi16 | tile_dim0
               (int)TILE1,                                     // tile_dim1 (tile_dim2=0)
               (int)(STRIDE0 & 0xffffffffull),                 // dim0_stride lo32
               (int)((STRIDE0 >> 32) & 0xffffull),             // dim0_stride hi16
               0 };
    v4i g2 = { 0, 0, 0, 0 };
    v4i g3 = { 0, 0, 0, 0 };
    v8i gx = { 0, 0, 0, 0, 0, 0, 0, 0 };
    __builtin_amdgcn_tensor_load_to_lds(g0, g1, g2, g3, gx, 0);
    wait_tensor0();
  }
#else
  // fallback: per-lane async copies of the 4 box rows
  #pragma unroll
  for (int a = 0; a < 4; ++a)
    for (int i4 = tid; i4 < PRE / 4; i4 += 1024)
      async_copy_b128(gbase + (size_t)a * CAND_STRIDE + i4 * 4, stage + a * PRE + i4 * 4);
#endif

  // ---- async-to-LDS: score row (ASYNCcnt path) ----
  for (int i4 = tid; i4 < PRE / 4; i4 += 1024)
    async_copy_b128(gbase + (size_t)4 * CAND_STRIDE + i4 * 4, sS + i4 * 4);
  wait_async0();
  __syncthreads();

  // ---- register-resident candidate set: item i at thread (i&1023), slot (i>>10) ----
  float y1[6], x1[6], y2[6], x2[6], sc[6], ar[6];
  #pragma unroll
  for (int k = 0; k < 6; ++k) {
    const int i = tid + k * 1024;
    if (i < PRE) {
      y1[k] = stage[0 * PRE + i]; x1[k] = stage[1 * PRE + i];
      y2[k] = stage[2 * PRE + i]; x2[k] = stage[3 * PRE + i];
      sc[k] = sS[i];
    } else { y1[k] = x1[k] = y2[k] = x2[k] = 0.f; sc[k] = NEG_INF; }
    ar[k] = (y2[k] - y1[k]) * (x2[k] - x1[k]);
  }
  __syncthreads();

  for (int it = 0; it < PROP; ++it) {
    float bs = NEG_INF; int bi = 0x7fffffff;
    #pragma unroll
    for (int k = 0; k < 6; ++k) {
      const int i = tid + k * 1024;
      if (sc[k] > bs || (sc[k] == bs && i < bi)) { bs = sc[k]; bi = i; }
    }
    #pragma unroll
    for (int off = 16; off > 0; off >>= 1) {
      const float os = __shfl_down(bs, off);
      const int   oi = __shfl_down(bi, off);
      if (os > bs || (os == bs && oi < bi)) { bs = os; bi = oi; }
    }
    if (lane == 0) { rS[wav] = bs; rI[wav] = bi; }
    __syncthreads();
    if (wav == 0) {
      bs = rS[lane]; bi = rI[lane];
      #pragma unroll
      for (int off = 16; off > 0; off >>= 1) {
        const float os = __shfl_down(bs, off);
        const int   oi = __shfl_down(bi, off);
        if (os > bs || (os == bs && oi < bi)) { bs = os; bi = oi; }
      }
      if (lane == 0) { bestS_sh = bs; bestI_sh = bi; }
    }
    __syncthreads();
    const float gbs = bestS_sh;
    const int   gbi = bestI_sh;
    if (tid == (gbi & 1023)) {
      const int k = gbi >> 10;
      bestBox[0] = y1[k]; bestBox[1] = x1[k]; bestBox[2] = y2[k]; bestBox[3] = x2[k];
      sc[k] = NEG_INF;
    }
    __syncthreads();

    const bool valid = gbs > NEG_INF;
    const float by1 = bestBox[0], bx1 = bestBox[1], by2 = bestBox[2], bx2 = bestBox[3];
    const float barea = (by2 - by1) * (bx2 - bx1);
    if (valid) {
      #pragma unroll
      for (int k = 0; k < 6; ++k) {
        const float iy1 = fmaxf(y1[k], by1), ix1 = fmaxf(x1[k], bx1);
        const float iy2 = fminf(y2[k], by2), ix2 = fminf(x2[k], bx2);
        const float inter = fmaxf(iy2 - iy1, 0.f) * fmaxf(ix2 - ix1, 0.f);
        const float uni = fmaxf(ar[k] + barea - inter, 1e-12f);
        if (inter / uni > NMS_T) sc[k] = NEG_INF;
      }
    }
    if (tid == 0) {
      const float4 o = valid ? make_float4(by1, bx1, by2, bx2)
                             : make_float4(0.f, 0.f, 0.f, 0.f);
      *(float4*)(out + ((long long)b * PROP + it) * 4) = o;
    }
  }
}

// =====================================================================
extern "C" void kernel_launch(void* const* d_in, const int* in_sizes, int n_in,
                              void* d_out, int out_size, void* d_ws, size_t ws_size,
                              hipStream_t stream) {
  const float* probs   = (const float*)d_in[0];   // (8, 262144, 2)
  const float* bbox    = (const float*)d_in[1];   // (8, 262144, 4)
  const float* anchors = (const float*)d_in[2];   // (8, 262144, 4)
  float* out = (float*)d_out;                     // (8, 1000, 4)

  int*   hist = (int*)d_ws;
  int*   meta = hist + HIST_INTS;
  float* cand = (float*)d_ws + CAND_OFF;

  (void)hipMemsetAsync(d_ws, 0, (size_t)(HIST_INTS + META_INTS) * sizeof(int), stream);

  hist_kernel   <<<(BATCH * NANCH) / 4096, 256, 0, stream>>>(probs, hist);
  thresh_kernel <<<BATCH, 1024, 0, stream>>>(hist, meta);
  compact_kernel<<<BATCH, 1024, 0, stream>>>(probs, bbox, anchors, meta, cand);
  nms_kernel    <<<BATCH, 1024, 0, stream>>>(cand, out);
}